// DIGNN_RW_62423054680273
// MI455X (gfx1250) — compile-verified
//
#include <hip/hip_runtime.h>
#include <math.h>

typedef float v2f __attribute__((ext_vector_type(2)));
typedef float v8f __attribute__((ext_vector_type(8)));

#define NN    100000
#define EE    3200000
#define INCH  256
#define HIDC  128
#define OUTC  47
#define MUF   1.0f
#define SCALE 0.5f          /* MU/(1+MU) */
#define INVP  0.5f          /* 1/(1+MU)  */
#define THRESHF 1e-6f
#define BNEPS 1e-5f
#define MAXIT 50
#define PFD   4             /* gather prefetch distance */

// ---------------- CSR construction ----------------

__global__ __launch_bounds__(256) void deg_count_k(const int* __restrict__ row,
                                                   int* __restrict__ deg) {
    int e = blockIdx.x * 256 + threadIdx.x;
    if (e < EE) atomicAdd(&deg[row[e]], 1);
}

// single-block exclusive scan over deg -> rowptr, cursor; also coef = (1/max(deg,1))/(1+mu)
__global__ __launch_bounds__(1024) void scan_k(const int* __restrict__ deg,
                                               int* __restrict__ rowptr,
                                               int* __restrict__ cursor,
                                               float* __restrict__ coef) {
    __shared__ int sh[1024];
    int tid = threadIdx.x;
    int carry = 0;
    for (int base = 0; base < NN; base += 1024) {
        int idx = base + tid;
        int d = (idx < NN) ? deg[idx] : 0;
        sh[tid] = d;
        __syncthreads();
        for (int off = 1; off < 1024; off <<= 1) {
            int t = (tid >= off) ? sh[tid - off] : 0;
            __syncthreads();
            sh[tid] += t;
            __syncthreads();
        }
        if (idx < NN) {
            int excl = carry + sh[tid] - d;
            rowptr[idx] = excl;
            cursor[idx] = excl;
            int dc = d > 1 ? d : 1;
            coef[idx] = INVP / (float)dc;
        }
        carry += sh[1023];
        __syncthreads();
    }
    if (tid == 0) rowptr[NN] = carry;
}

__global__ __launch_bounds__(256) void csr_fill_k(const int* __restrict__ row,
                                                  const int* __restrict__ col,
                                                  int* __restrict__ cursor,
                                                  int* __restrict__ csrcol) {
    int e = blockIdx.x * 256 + threadIdx.x;
    if (e < EE) {
        int r = row[e];
        int p = atomicAdd(&cursor[r], 1);
        csrcol[p] = col[e];
    }
}

// ---------------- ax = A @ x  (gather, 256 ch, wave per node) ----------------

__global__ __launch_bounds__(256) void agg_x_k(const int* __restrict__ rowptr,
                                               const int* __restrict__ csrcol,
                                               const float* __restrict__ x,
                                               float* __restrict__ ax) {
    int node = blockIdx.x * 8 + (threadIdx.x >> 5);
    int lane = threadIdx.x & 31;
    int s = rowptr[node], e = rowptr[node + 1];
    int off0 = lane * 4, off1 = 128 + lane * 4;
    float4 a0 = {0.f, 0.f, 0.f, 0.f};
    float4 a1 = {0.f, 0.f, 0.f, 0.f};
    for (int i = s; i < e; ++i) {
        if (i + PFD < e) {
            int cp = csrcol[i + PFD];
            __builtin_prefetch(x + (size_t)cp * INCH + off0, 0, 3);
            __builtin_prefetch(x + (size_t)cp * INCH + off1, 0, 3);
        }
        int c = csrcol[i];
        const float* xr = x + (size_t)c * INCH;
        float4 v0 = *(const float4*)(xr + off0);
        float4 v1 = *(const float4*)(xr + off1);
        a0.x += v0.x; a0.y += v0.y; a0.z += v0.z; a0.w += v0.w;
        a1.x += v1.x; a1.y += v1.y; a1.z += v1.z; a1.w += v1.w;
    }
    float* orow = ax + (size_t)node * INCH;
    *(float4*)(orow + off0) = a0;
    *(float4*)(orow + off1) = a1;
}

// ---------------- fc1 + relu + BN + mu-scale via WMMA f32 16x16x4 ----------------

__global__ __launch_bounds__(256) void fc1_bn_k(const float* __restrict__ ax,
                                                const float* __restrict__ w1,
                                                const float* __restrict__ b1,
                                                const float* __restrict__ gamma,
                                                const float* __restrict__ beta,
                                                const float* __restrict__ rmean,
                                                const float* __restrict__ rvar,
                                                float* __restrict__ xs) {
    int wave = blockIdx.x * 8 + (threadIdx.x >> 5);   // 0..49999
    int lane = threadIdx.x & 31;
    int mtile = wave >> 3;       // 0..6249
    int ntile = wave & 7;        // 0..7
    int m0 = mtile * 16, n0 = ntile * 16;
    int h = lane >> 4, q = lane & 15;

    const float* arow = ax + (size_t)(m0 + q) * INCH;
    const float* brow = w1 + (size_t)(n0 + q) * INCH;

    v8f c = {};
    for (int k = 0; k < INCH; k += 4) {
        int kk = k + 2 * h;
        v2f a; a.x = arow[kk]; a.y = arow[kk + 1];
        v2f b; b.x = brow[kk]; b.y = brow[kk + 1];
        c = __builtin_amdgcn_wmma_f32_16x16x4_f32(false, a, false, b,
                                                  (short)0, c, false, false);
    }

    int col = n0 + q;
    float g  = gamma[col];
    float bt = beta[col];
    float mn = rmean[col];
    float iv = rsqrtf(rvar[col] + BNEPS);
    float bb = b1[col];
#pragma unroll
    for (int r = 0; r < 8; ++r) {
        int rowi = m0 + r + 8 * h;
        float v = c[r] + bb;
        v = fmaxf(v, 0.0f);
        v = g * (v - mn) * iv + bt;
        xs[(size_t)rowi * HIDC + col] = v * SCALE;
    }
}

// ---------------- Picard iteration: zn = coef * (A z) + xs ----------------

__global__ __launch_bounds__(256) void prop_k(const int* __restrict__ rowptr,
                                              const int* __restrict__ csrcol,
                                              const float* __restrict__ coef,
                                              const float* __restrict__ xs,
                                              const float* __restrict__ zin,
                                              float* __restrict__ zout,
                                              float* __restrict__ sums,
                                              const int* __restrict__ flags) {
    if (flags[0]) return;   // converged: no-op (deterministic per input data)
    int node = blockIdx.x * 8 + (threadIdx.x >> 5);
    int lane = threadIdx.x & 31;
    int s = rowptr[node], e = rowptr[node + 1];
    int off = lane * 4;
    float4 acc = {0.f, 0.f, 0.f, 0.f};
    for (int i = s; i < e; ++i) {
        if (i + PFD < e) {
            int cp = csrcol[i + PFD];
            __builtin_prefetch(zin + (size_t)cp * HIDC + off, 0, 3);
        }
        int c = csrcol[i];
        float4 zv = *(const float4*)(zin + (size_t)c * HIDC + off);
        acc.x += zv.x; acc.y += zv.y; acc.z += zv.z; acc.w += zv.w;
    }
    float cf = coef[node];
    float4 xv = *(const float4*)(xs + (size_t)node * HIDC + off);
    float4 zo = *(const float4*)(zin + (size_t)node * HIDC + off);
    float4 zn;
    zn.x = acc.x * cf + xv.x;
    zn.y = acc.y * cf + xv.y;
    zn.z = acc.z * cf + xv.z;
    zn.w = acc.w * cf + xv.w;
    *(float4*)(zout + (size_t)node * HIDC + off) = zn;

    float dx = zn.x - zo.x, dy = zn.y - zo.y, dz = zn.z - zo.z, dw = zn.w - zo.w;
    float d1 = dx * dx + dy * dy + dz * dz + dw * dw;
    float d2 = zo.x * zo.x + zo.y * zo.y + zo.z * zo.z + zo.w * zo.w;
    for (int m = 1; m < 32; m <<= 1) {
        d1 += __shfl_xor(d1, m, 32);
        d2 += __shfl_xor(d2, m, 32);
    }
    if (lane == 0) {
        atomicAdd(&sums[0], d1);
        atomicAdd(&sums[1], d2);
    }
}

__global__ void check_k(float* sums, int* flags, int out_idx) {
    if (threadIdx.x == 0 && blockIdx.x == 0) {
        if (!flags[0]) {
            float diff = sqrtf(sums[0]) / (sqrtf(sums[1]) + 1e-8f);
            flags[1] = out_idx;                 // buffer holding latest z
            if (diff <= THRESHF) flags[0] = 1;  // stop iterating
        }
        sums[0] = 0.f;
        sums[1] = 0.f;
    }
}

// ---------------- fc2 + log_softmax via WMMA f32 16x16x4 ----------------

__global__ __launch_bounds__(256) void fc2_lsm_k(const float* __restrict__ zb0,
                                                 const float* __restrict__ zb1,
                                                 const int* __restrict__ flags,
                                                 const float* __restrict__ w2,
                                                 const float* __restrict__ b2,
                                                 float* __restrict__ out) {
    int wave = blockIdx.x * 8 + (threadIdx.x >> 5);
    if (wave >= NN / 16) return;                 // uniform per wave
    int lane = threadIdx.x & 31;
    int h = lane >> 4, q = lane & 15;
    const float* z = flags[1] ? zb1 : zb0;
    int m0 = wave * 16;

    const float* arow = z + (size_t)(m0 + q) * HIDC;
    const float* br0  = w2 + (size_t)q * HIDC;          // cols 0..15
    const float* br1  = w2 + (size_t)(16 + q) * HIDC;   // cols 16..31
    int   n2valid = (32 + q) < OUTC;
    float wv = n2valid ? 1.f : 0.f;
    const float* br2  = w2 + (size_t)(n2valid ? (32 + q) : 0) * HIDC;

    v8f c0 = {}, c1 = {}, c2 = {};
    for (int k = 0; k < HIDC; k += 4) {
        int kk = k + 2 * h;
        v2f a; a.x = arow[kk]; a.y = arow[kk + 1];
        v2f b; b.x = br0[kk]; b.y = br0[kk + 1];
        c0 = __builtin_amdgcn_wmma_f32_16x16x4_f32(false, a, false, b, (short)0, c0, false, false);
        b.x = br1[kk]; b.y = br1[kk + 1];
        c1 = __builtin_amdgcn_wmma_f32_16x16x4_f32(false, a, false, b, (short)0, c1, false, false);
        b.x = br2[kk] * wv; b.y = br2[kk + 1] * wv;
        c2 = __builtin_amdgcn_wmma_f32_16x16x4_f32(false, a, false, b, (short)0, c2, false, false);
    }

    float bias0 = b2[q];
    float bias1 = b2[16 + q];
    float bias2 = n2valid ? b2[32 + q] : 0.f;

#pragma unroll
    for (int r = 0; r < 8; ++r) {
        float v0 = c0[r] + bias0;
        float v1 = c1[r] + bias1;
        float v2 = c2[r] + bias2;
        float v2m = n2valid ? v2 : -3.0e38f;
        float mx = fmaxf(fmaxf(v0, v1), v2m);
        for (int m = 1; m < 16; m <<= 1) mx = fmaxf(mx, __shfl_xor(mx, m, 32));
        float s = __expf(v0 - mx) + __expf(v1 - mx) + (n2valid ? __expf(v2 - mx) : 0.f);
        for (int m = 1; m < 16; m <<= 1) s += __shfl_xor(s, m, 32);
        float lse = __logf(s);
        int rowi = m0 + r + 8 * h;
        float* orow = out + (size_t)rowi * OUTC;
        orow[q]      = v0 - mx - lse;
        orow[16 + q] = v1 - mx - lse;
        if (n2valid) orow[32 + q] = v2 - mx - lse;
    }
}

// ---------------- launch ----------------

extern "C" void kernel_launch(void* const* d_in, const int* in_sizes, int n_in,
                              void* d_out, int out_size, void* d_ws, size_t ws_size,
                              hipStream_t stream) {
    (void)in_sizes; (void)n_in; (void)out_size; (void)ws_size;

    const float* x     = (const float*)d_in[0];
    const int*   ei    = (const int*)d_in[1];
    const float* w1    = (const float*)d_in[2];
    const float* b1    = (const float*)d_in[3];
    const float* gamma = (const float*)d_in[4];
    const float* beta  = (const float*)d_in[5];
    const float* rmean = (const float*)d_in[6];
    const float* rvar  = (const float*)d_in[7];
    const float* w2    = (const float*)d_in[8];
    const float* b2    = (const float*)d_in[9];
    const int* row = ei;
    const int* col = ei + EE;

    char* ws = (char*)d_ws;
    size_t off = 0;
    auto alloc = [&](size_t bytes) -> char* {
        char* p = ws + off;
        off = (off + bytes + 255) & ~(size_t)255;
        return p;
    };
    float* ax     = (float*)alloc((size_t)NN * INCH * 4);
    float* xs     = (float*)alloc((size_t)NN * HIDC * 4);
    float* z0     = (float*)alloc((size_t)NN * HIDC * 4);
    float* z1     = (float*)alloc((size_t)NN * HIDC * 4);
    float* coef   = (float*)alloc((size_t)NN * 4);
    int*   rowptr = (int*)alloc((size_t)(NN + 1) * 4);
    int*   cursor = (int*)alloc((size_t)NN * 4);
    int*   degint = (int*)alloc((size_t)NN * 4);
    int*   csrcol = (int*)alloc((size_t)EE * 4);
    float* sums   = (float*)alloc(2 * 4);
    int*   flags  = (int*)alloc(2 * 4);

    hipMemsetAsync(degint, 0, (size_t)NN * 4, stream);
    hipMemsetAsync(z0, 0, (size_t)NN * HIDC * 4, stream);
    hipMemsetAsync(sums, 0, 8, stream);
    hipMemsetAsync(flags, 0, 8, stream);

    deg_count_k<<<(EE + 255) / 256, 256, 0, stream>>>(row, degint);
    scan_k<<<1, 1024, 0, stream>>>(degint, rowptr, cursor, coef);
    csr_fill_k<<<(EE + 255) / 256, 256, 0, stream>>>(row, col, cursor, csrcol);
    agg_x_k<<<NN / 8, 256, 0, stream>>>(rowptr, csrcol, x, ax);

    // 50000 waves (6250 M-tiles x 8 N-tiles), 8 waves per 256-thread block -> 6250 blocks
    fc1_bn_k<<<NN / 16, 256, 0, stream>>>(ax, w1, b1, gamma, beta, rmean, rvar, xs);

    for (int it = 0; it < MAXIT; ++it) {
        const float* zi = (it & 1) ? z1 : z0;
        float*       zo = (it & 1) ? z0 : z1;
        prop_k<<<NN / 8, 256, 0, stream>>>(rowptr, csrcol, coef, xs, zi, zo, sums, flags);
        check_k<<<1, 64, 0, stream>>>(sums, flags, 1 - (it & 1));
    }

    fc2_lsm_k<<<(NN / 16 + 7) / 8, 256, 0, stream>>>(z0, z1, flags, w2, b2, (float*)d_out);
}